// GCNConv_1554778161396
// MI455X (gfx1250) — compile-verified
//
#include <hip/hip_runtime.h>

typedef __attribute__((ext_vector_type(2))) float v2f;
typedef __attribute__((ext_vector_type(8))) float v8f;

#define IN_DIM 128
#define OUT_DIM 128

// -------------------------------------------------------------------------
// Kernel 1: h = x @ w via V_WMMA_F32_16X16X4_F32 (full f32 precision).
// 256 threads = 8 waves; each wave owns one 16-row slab of x and computes
// the full 16x128 output slab (8 N-tiles, 32 K-steps of K=4).
// w (64 KB) is staged in LDS once per block.
// -------------------------------------------------------------------------
__global__ __launch_bounds__(256) void gemm_xw_wmma(
    const float* __restrict__ x, const float* __restrict__ w,
    float* __restrict__ h, int n_nodes) {
  __shared__ float wlds[IN_DIM * OUT_DIM];  // 64 KB of the 320 KB WGP LDS
  for (int i = threadIdx.x; i < IN_DIM * OUT_DIM; i += 256)
    wlds[i] = w[i];
  __syncthreads();

  const int lane = threadIdx.x & 31;
  const int wave = threadIdx.x >> 5;
  const int tile = blockIdx.x * 8 + wave;  // 16-row tile index
  const int row0 = tile * 16;
  if (row0 >= n_nodes) return;  // wave-uniform: EXEC stays all-1s for WMMA

  const int m  = lane & 15;        // M (for A) / N (for B,C) position
  const int kk = (lane >> 4) * 2;  // lane-half K offset: 0 or 2

  v8f acc[8];
  #pragma unroll
  for (int t = 0; t < 8; ++t) acc[t] = (v8f){};

  for (int k0 = 0; k0 < IN_DIM; k0 += 4) {
    // A fragment: x[row0+m][k0+kk .. k0+kk+1]  (clamped for ragged tail)
    int arow = row0 + m;
    if (arow >= n_nodes) arow = n_nodes - 1;
    const v2f a = *(const v2f*)(x + (size_t)arow * IN_DIM + (k0 + kk));

    #pragma unroll
    for (int t = 0; t < 8; ++t) {
      const int n0 = t * 16;
      v2f bf;
      bf.x = wlds[(k0 + kk)     * OUT_DIM + n0 + m];
      bf.y = wlds[(k0 + kk + 1) * OUT_DIM + n0 + m];
      acc[t] = __builtin_amdgcn_wmma_f32_16x16x4_f32(
          /*neg_a=*/false, a, /*neg_b=*/false, bf,
          /*c_mod=*/(short)0, acc[t], /*reuse_a=*/false, /*reuse_b=*/false);
    }
  }

  // C/D layout: VGPR r -> row (lane<16 ? r : r+8), col = lane%16 + 16*t
  const int mbase = row0 + (lane >> 4) * 8;
  #pragma unroll
  for (int t = 0; t < 8; ++t) {
    const int n = (lane & 15) + t * 16;
    #pragma unroll
    for (int r = 0; r < 8; ++r) {
      const int row = mbase + r;
      if (row < n_nodes) h[(size_t)row * OUT_DIM + n] = acc[t][r];
    }
  }
}

// -------------------------------------------------------------------------
// Kernel 2: out[i][j] = b[j]  (bias init; also zeroes the poisoned buffer)
// -------------------------------------------------------------------------
__global__ __launch_bounds__(256) void init_out_bias(
    const float* __restrict__ b, float* __restrict__ out, int n) {
  const int i = blockIdx.x * 256 + threadIdx.x;
  if (i < n) out[i] = b[i & (OUT_DIM - 1)];
}

// -------------------------------------------------------------------------
// Kernel 3: SpMM scatter.  One wave per edge; lane L handles columns
// 4L..4L+3 via a float4 gather from h[col] (L2-resident: 51 MB < 192 MB)
// and 4 global_atomic_add_f32 into out[row].
// -------------------------------------------------------------------------
__global__ __launch_bounds__(256) void spmm_scatter(
    const float* __restrict__ h, const float* __restrict__ ew,
    const int* __restrict__ erow, const int* __restrict__ ecol,
    float* __restrict__ out, int n_edges) {
  const int lane = threadIdx.x & 31;
  const int e = blockIdx.x * 8 + (threadIdx.x >> 5);
  if (e >= n_edges) return;

  const int   src = ecol[e];
  const int   dst = erow[e];
  const float wgt = ew[e];

  const float4 v = *(const float4*)(h + (size_t)src * OUT_DIM + lane * 4);
  float* o = out + (size_t)dst * OUT_DIM + lane * 4;
  atomicAdd(o + 0, wgt * v.x);
  atomicAdd(o + 1, wgt * v.y);
  atomicAdd(o + 2, wgt * v.z);
  atomicAdd(o + 3, wgt * v.w);
}

// -------------------------------------------------------------------------
extern "C" void kernel_launch(void* const* d_in, const int* in_sizes, int n_in,
                              void* d_out, int out_size, void* d_ws, size_t ws_size,
                              hipStream_t stream) {
  const float* x    = (const float*)d_in[0];
  const float* w    = (const float*)d_in[1];
  const float* b    = (const float*)d_in[2];
  const float* ew   = (const float*)d_in[3];
  const int*   erow = (const int*)d_in[4];
  const int*   ecol = (const int*)d_in[5];
  float* out = (float*)d_out;
  float* h   = (float*)d_ws;  // [n_nodes, 128] f32 scratch

  const int n_nodes = in_sizes[0] / IN_DIM;
  const int n_edges = in_sizes[3];

  const int n_tiles = (n_nodes + 15) / 16;
  gemm_xw_wmma<<<dim3((n_tiles + 7) / 8), dim3(256), 0, stream>>>(x, w, h, n_nodes);

  const int n_out = n_nodes * OUT_DIM;
  init_out_bias<<<dim3((n_out + 255) / 256), dim3(256), 0, stream>>>(b, out, n_out);

  spmm_scatter<<<dim3((n_edges + 7) / 8), dim3(256), 0, stream>>>(
      h, ew, erow, ecol, out, n_edges);
}